// GraphReconstructionLoss_48842368090167
// MI455X (gfx1250) — compile-verified
//
#include <hip/hip_runtime.h>
#include <hip/hip_bf16.h>
#include <math.h>

// ---------------------------------------------------------------------------
// GraphReconstructionLoss on MI455X (gfx1250, wave32)
//
// score(e) = dot(emb[src[e]], emb[dst[e]]), D=512, fp32.
// loss = mean over pos/neg of softplus(+-score).
//
// Strategy: one wave32 handles 16 edges. Accumulate C += A x B with
// V_WMMA_F32_16X16X4_F32 over 128 K-chunks of 4; diag(C) holds the 16 dot
// products in exact fp32. Table (102.4 MB) fits in the 192 MB L2, so the
// 1.2 GB of gathered row reads are L2-resident -> L2-bandwidth bound.
// ---------------------------------------------------------------------------

typedef __attribute__((ext_vector_type(2))) float v2f;
typedef __attribute__((ext_vector_type(8))) float v8f;

static constexpr int D    = 512;  // embed dim (fixed by reference)
static constexpr int TILE = 16;   // edges per wave-tile

__device__ __forceinline__ float softplus_stable(float x) {
    // log(1 + exp(x)) = max(x,0) + log1p(exp(-|x|))
    return fmaxf(x, 0.0f) + log1pf(expf(-fabsf(x)));
}

__global__ void __launch_bounds__(256)
edge_loss_partial(const float* __restrict__ emb,
                  const int*   __restrict__ pos_edges,  // [2,E] flat: src | dst
                  const int*   __restrict__ neg_edges,  // [2,E] flat
                  int E, int nPosTiles,
                  float* __restrict__ partial)
{
    const int lane = threadIdx.x & 31;
    const int wave = threadIdx.x >> 5;
    const int tile = blockIdx.x * (blockDim.x >> 5) + wave;   // wave-uniform
    const int nTiles = 2 * nPosTiles;

    float laneVal = 0.0f;

    if (tile < nTiles) {          // wave-uniform branch: EXEC stays all-1s
        const bool isPos = (tile < nPosTiles);
        const int  tbase = (isPos ? tile : tile - nPosTiles) * TILE;
        const int* edges = isPos ? pos_edges : neg_edges;

        // A-tile layout (16x4 f32): lanes 0-15 own row=lane, K = k0,k0+1 (VGPR0,1);
        // lanes 16-31 own row=lane-16, K = k0+2,k0+3. B mirrors with dst rows.
        const int row    = lane & 15;
        int e            = tbase + row;
        const int eClamp = (e < E) ? e : (E - 1);             // safe gather
        const int srcRow = edges[eClamp];
        const int dstRow = edges[(size_t)E + eClamp];
        const int kHalf  = (lane >> 4) << 1;                  // 0 or 2

        const float* pa = emb + (size_t)srcRow * D + kHalf;
        const float* pb = emb + (size_t)dstRow * D + kHalf;

        v8f c = {};  // 16x16 f32 accumulator (8 VGPRs)
#pragma unroll 8
        for (int k0 = 0; k0 < D; k0 += 4) {
            v2f a = *(const v2f*)(pa + k0);   // global_load_b64, L2-resident
            v2f b = *(const v2f*)(pb + k0);
            // D = A(16x4) x B(4x16) + C ; fp32-exact matrix op
            c = __builtin_amdgcn_wmma_f32_16x16x4_f32(
                    false, a, false, b, (short)0, c, false, false);
        }

        // diag(C): element (i,i) -> lane i, VGPR i      (i = 0..7)
        //                         -> lane 16+i, VGPR i-8 (i = 8..15)
        const bool active = (lane < 8) || (lane >= 24);
        const int  sel    = lane & 7;
        float s = 0.0f;
#pragma unroll
        for (int i = 0; i < 8; ++i)
            if (sel == i) s = c[i];

        const int myRow  = (lane < 8) ? lane : (lane - 16);   // diag index
        const int myEdge = tbase + myRow;
        if (active && myEdge < E)
            laneVal = isPos ? softplus_stable(-s) : softplus_stable(s);
    }

    // Deterministic wave32 reduction
#pragma unroll
    for (int off = 16; off > 0; off >>= 1)
        laneVal += __shfl_xor(laneVal, off, 32);

    __shared__ float lds[8];
    if (lane == 0) lds[wave] = laneVal;
    __syncthreads();
    if (threadIdx.x == 0) {
        float bsum = 0.0f;
        const int nw = blockDim.x >> 5;
        for (int w = 0; w < nw; ++w) bsum += lds[w];          // fixed order
        partial[blockIdx.x] = bsum;
    }
}

__global__ void __launch_bounds__(256)
reduce_final(const float* __restrict__ partial, int n, float invCount,
             float* __restrict__ out)
{
    __shared__ float lds[256];
    float s = 0.0f;
    for (int i = threadIdx.x; i < n; i += 256) s += partial[i];  // fixed order
    lds[threadIdx.x] = s;
    __syncthreads();
#pragma unroll
    for (int stride = 128; stride > 0; stride >>= 1) {
        if ((int)threadIdx.x < stride) lds[threadIdx.x] += lds[threadIdx.x + stride];
        __syncthreads();
    }
    if (threadIdx.x == 0) out[0] = lds[0] * invCount;
}

extern "C" void kernel_launch(void* const* d_in, const int* in_sizes, int n_in,
                              void* d_out, int out_size, void* d_ws, size_t ws_size,
                              hipStream_t stream)
{
    const float* emb = (const float*)d_in[0];   // [N_NODES, 512] fp32
    const int*   pos = (const int*)d_in[1];     // [2, E] int
    const int*   neg = (const int*)d_in[2];     // [2, E] int

    const int E         = in_sizes[1] / 2;
    const int nPosTiles = (E + TILE - 1) / TILE;
    const int nTiles    = 2 * nPosTiles;

    const int wavesPerBlock = 8;                // 256 threads = 8 wave32
    const int blocks = (nTiles + wavesPerBlock - 1) / wavesPerBlock;

    float* partial = (float*)d_ws;              // blocks * 4 bytes of scratch

    edge_loss_partial<<<blocks, 256, 0, stream>>>(emb, pos, neg, E, nPosTiles, partial);
    reduce_final<<<1, 256, 0, stream>>>(partial, blocks, 1.0f / (2.0f * (float)E),
                                        (float*)d_out);
}